// SinRegPrediction_51153060495816
// MI455X (gfx1250) — compile-verified
//
#include <hip/hip_runtime.h>

#define X_DIM 64
#define N_FREQ 16
#define QPB 256          // q-values (= threads) per block

// y[b,q] = sum_{d,f} amp[b,d,f] * sin(x_q[b,q,d] * freq[b,d,f])
//   z   : [B, 2*X_DIM*N_FREQ]  (first half amplitudes, second half freqs)
//   x_q : [B, Q, X_DIM]
//   out : [B, Q]
// Grid: x = Q/QPB q-tiles, y = B batches  (no integer division in kernel)
__global__ __launch_bounds__(QPB) void sinreg_kernel(
    const float* __restrict__ z,
    const float* __restrict__ x_q,
    float* __restrict__ out,
    int q_total)
{
    extern __shared__ float smem[];           // [0,1024)=amp  [1024,2048)=freq  (8 KB)
    const int t = threadIdx.x;
    const int b = blockIdx.y;
    const int q = blockIdx.x * QPB + t;

    const float* zb = z + (size_t)b * (2 * X_DIM * N_FREQ);

    // ---- CDNA5 async copy: stage z[b] (8 KB) into LDS --------------------
    // 2048 floats = 512 b128 transfers; 256 lanes issue 2 each.
    // Dynamic LDS with zero static LDS => LDS byte offset == array byte index.
    {
        unsigned off0 = (unsigned)t * 16u;          // bytes, both LDS and global
        unsigned off1 = off0 + 256u * 16u;
        asm volatile(
            "global_load_async_to_lds_b128 %0, %1, %2\n\t"
            "global_load_async_to_lds_b128 %3, %4, %2"
            :
            : "v"(off0), "v"(off0), "s"(zb), "v"(off1), "v"(off1)
            : "memory");
    }

    const float* xrow = x_q + ((size_t)b * q_total + q) * X_DIM;
    __builtin_prefetch(xrow, 0, 1);   // global_prefetch_b8: warm this lane's x row

    asm volatile("s_wait_asynccnt 0x0" ::: "memory");
    __syncthreads();

    const float* sAmp  = smem;
    const float* sFreq = smem + X_DIM * N_FREQ;

    float acc = 0.0f;
    const float4* xr4 = (const float4*)xrow;

#pragma unroll 2
    for (int c = 0; c < X_DIM / 4; ++c) {     // 16 trips, 4 d's each; 2x unrolled
        float4 xv = xr4[c];
        float xs[4] = {xv.x, xv.y, xv.z, xv.w};
#pragma unroll
        for (int j = 0; j < 4; ++j) {
            const int d = c * 4 + j;
            const float x = xs[j];
            // wave-uniform LDS addresses -> conflict-free broadcast reads
            const float4* a4 = (const float4*)(sAmp  + d * N_FREQ);
            const float4* f4 = (const float4*)(sFreq + d * N_FREQ);
#pragma unroll
            for (int g = 0; g < 4; ++g) {
                float4 a  = a4[g];
                float4 fr = f4[g];
                acc += a.x * __sinf(x * fr.x);   // v_sin_f32 (TRANS) co-executes
                acc += a.y * __sinf(x * fr.y);   // with the v_fmac_f32 VALU work
                acc += a.z * __sinf(x * fr.z);
                acc += a.w * __sinf(x * fr.w);
            }
        }
    }

    out[(size_t)b * q_total + q] = acc;
}

extern "C" void kernel_launch(void* const* d_in, const int* in_sizes, int n_in,
                              void* d_out, int out_size, void* d_ws, size_t ws_size,
                              hipStream_t stream) {
    const float* z   = (const float*)d_in[0];   // [B, 2048]
    const float* x_q = (const float*)d_in[1];   // [B, Q, 64]
    float* out = (float*)d_out;                 // [B, Q, 1]

    const int B = in_sizes[0] / (2 * X_DIM * N_FREQ);   // 32
    const int Q = in_sizes[1] / (B * X_DIM);            // 4096

    dim3 grid((unsigned)(Q / QPB), (unsigned)B);        // 16 x 32 = 512 blocks
    dim3 block(QPB);                                    // 256 threads = 8 wave32
    size_t lds_bytes = 2 * X_DIM * N_FREQ * sizeof(float);  // 8192

    sinreg_kernel<<<grid, block, lds_bytes, stream>>>(z, x_q, out, Q);
}